// lowRNN_75093208203469
// MI455X (gfx1250) — compile-verified
//
#include <hip/hip_runtime.h>
#include <math.h>

// ---------------- problem constants ----------------
#define NN   1024   // neurons
#define RR   8      // rank
#define INC  8      // input channels
#define OUTC 4      // output channels
#define BB   64     // batch
#define TT   1000   // time steps

constexpr float ALPHA_C  = 10.0f / 100.0f;            // DT/TAU = 0.1
constexpr float OMALPHA  = 1.0f - ALPHA_C;            // 0.9
constexpr float NOISE_SC = 1.4142135623730951f;       // sqrt(2*TAU)*G_REC
constexpr float INV_N    = 1.0f / (float)NN;

typedef float v2f __attribute__((ext_vector_type(2)));
typedef float v8f __attribute__((ext_vector_type(8)));

__device__ __forceinline__ float my_tanh(float x) {
#if __has_builtin(__builtin_amdgcn_tanhf)
    return __builtin_amdgcn_tanhf(x);   // v_tanh_f32 on gfx1250
#else
    return tanhf(x);
#endif
}

// ================= scan kernel =================
// One workgroup per batch element. 256 threads, 4 neurons/thread.
// All weights register-resident; input series in LDS; noise prefetched.
__global__ __launch_bounds__(256, 1)
void rnn_scan_kernel(const float* __restrict__ X,      // [B,T,INC]
                     const float* __restrict__ noise,  // [B,T,N]
                     const float* __restrict__ Win,    // [N,INC]
                     const float* __restrict__ WV,     // [R,N]
                     const float* __restrict__ WU,     // [N,R]
                     const float* __restrict__ s_in,   // [INC]
                     const float* __restrict__ h0,     // [N]
                     float* __restrict__ hidden)       // [B,T,N]
{
    __shared__ float xl[TT * INC];       // 32000 B : scaled input series
    __shared__ float part[RR * 260];     //  8320 B : per-thread rank partials
    __shared__ float red2[64];           // segment partial sums
    __shared__ float finals[RR];         // reduced v[r]

    const int b   = blockIdx.x;
    const int tid = threadIdx.x;
    const int n0  = tid * 4;

    // ---- preload scaled input series into LDS ----
    for (int i = tid; i < TT * INC; i += 256)
        xl[i] = X[(size_t)b * TT * INC + i] * s_in[i & (INC - 1)];

    // ---- register-resident weights for this thread's 4 neurons ----
    float wv[RR][4], wu[4][RR], wi[4][INC], h[4];
#pragma unroll
    for (int r = 0; r < RR; ++r) {
        float4 w = *(const float4*)(WV + (size_t)r * NN + n0);
        wv[r][0] = w.x; wv[r][1] = w.y; wv[r][2] = w.z; wv[r][3] = w.w;
    }
#pragma unroll
    for (int j = 0; j < 4; ++j) {
        float4 a = *(const float4*)(WU + (size_t)(n0 + j) * RR);
        float4 c = *(const float4*)(WU + (size_t)(n0 + j) * RR + 4);
        wu[j][0] = a.x; wu[j][1] = a.y; wu[j][2] = a.z; wu[j][3] = a.w;
        wu[j][4] = c.x; wu[j][5] = c.y; wu[j][6] = c.z; wu[j][7] = c.w;
        float4 p = *(const float4*)(Win + (size_t)(n0 + j) * INC);
        float4 q = *(const float4*)(Win + (size_t)(n0 + j) * INC + 4);
        wi[j][0] = p.x; wi[j][1] = p.y; wi[j][2] = p.z; wi[j][3] = p.w;
        wi[j][4] = q.x; wi[j][5] = q.y; wi[j][6] = q.z; wi[j][7] = q.w;
    }
    {
        float4 hh = *(const float4*)(h0 + n0);
        h[0] = hh.x; h[1] = hh.y; h[2] = hh.z; h[3] = hh.w;
    }

    __syncthreads();

    const size_t baseBT = (size_t)b * TT;
    float4 nzA = *(const float4*)(noise + baseBT * NN + n0);  // t = 0
    float4 nzB = nzA;

    for (int t = 0; t < TT; ++t) {
        // prefetch next step's noise early (hide global latency behind reduce)
        if (t + 1 < TT)
            nzB = *(const float4*)(noise + (baseBT + t + 1) * NN + n0);

        const float a0 = my_tanh(h[0]);
        const float a1 = my_tanh(h[1]);
        const float a2 = my_tanh(h[2]);
        const float a3 = my_tanh(h[3]);

        // rank partials p[r] = sum over this thread's 4 neurons
#pragma unroll
        for (int r = 0; r < RR; ++r) {
            float p = a0 * wv[r][0];
            p = fmaf(a1, wv[r][1], p);
            p = fmaf(a2, wv[r][2], p);
            p = fmaf(a3, wv[r][3], p);
            part[r * 260 + tid] = p;
        }
        __syncthreads();

        // phase 2: 64 threads, each sums 32 partials of one rank (float4 loads)
        if (tid < 64) {
            const int val = tid >> 3, seg = tid & 7;
            const float4* pp = (const float4*)&part[val * 260 + seg * 32];
            float s0 = 0.f, s1 = 0.f, s2 = 0.f, s3 = 0.f;
#pragma unroll
            for (int k = 0; k < 8; ++k) {
                float4 v4 = pp[k];
                s0 += v4.x; s1 += v4.y; s2 += v4.z; s3 += v4.w;
            }
            red2[tid] = (s0 + s1) + (s2 + s3);
        }
        __syncthreads();

        // phase 3: 8 threads finish each rank
        if (tid < RR) {
            float s = 0.f;
#pragma unroll
            for (int k = 0; k < 8; ++k) s += red2[tid * 8 + k];
            finals[tid] = s;
        }
        __syncthreads();

        float v[RR];
#pragma unroll
        for (int r = 0; r < RR; ++r) v[r] = finals[r];  // LDS broadcast

        // per-step input vector: two broadcast b128 loads
        const float4 x0 = *(const float4*)&xl[t * INC];
        const float4 x1 = *(const float4*)&xl[t * INC + 4];
        const float xc[INC] = {x0.x, x0.y, x0.z, x0.w, x1.x, x1.y, x1.z, x1.w};

        float hn[4];
#pragma unroll
        for (int j = 0; j < 4; ++j) {
            float rec = 0.f;
#pragma unroll
            for (int r = 0; r < RR; ++r) rec = fmaf(v[r], wu[j][r], rec);
            float u = 0.f;
#pragma unroll
            for (int c = 0; c < INC; ++c) u = fmaf(xc[c], wi[j][c], u);
            float nz = (j == 0) ? nzA.x : (j == 1) ? nzA.y : (j == 2) ? nzA.z : nzA.w;
            hn[j] = OMALPHA * h[j] + ALPHA_C * (rec * INV_N + u + NOISE_SC * nz);
        }
        h[0] = hn[0]; h[1] = hn[1]; h[2] = hn[2]; h[3] = hn[3];

        *(float4*)(hidden + (baseBT + t) * NN + n0) =
            make_float4(h[0], h[1], h[2], h[3]);

        nzA = nzB;
    }
}

// ================= WMMA readout kernel =================
// out[m, c] = sum_k tanh(H[m,k]) * (W_out[c,k] * s_out[c] / N),  m = b*T+t
// GEMM M=64000, K=1024, Nout=4 (padded to 16). One 16-row tile per wave.
// V_WMMA_F32_16X16X4_F32: A 16x4 f32 (2 VGPR), B 4x16 f32 (2 VGPR), C/D 16x16 f32.
// B lives in LDS: 4 scaled W_out rows + one zero row for padded columns,
// row stride 1028 floats so rows hit distinct bank quads (branch-free b64 loads).
#define NP (NN + 4)

__global__ __launch_bounds__(256, 2)
void readout_wmma_kernel(const float* __restrict__ H,     // [B*T, N]
                         const float* __restrict__ Wout,  // [OUTC, N]
                         const float* __restrict__ s_out, // [OUTC]
                         float* __restrict__ out)         // [B*T, OUTC]
{
    __shared__ float wl[5 * NP];   // rows 0..3 scaled W_out, row 4 zeros (~20.6 KB)
    const int tid = threadIdx.x;
    for (int i = tid; i < 5 * NP; i += 256) {
        const int row = i / NP, k = i - row * NP;
        float vv = 0.0f;
        if (row < OUTC && k < NN)
            vv = Wout[row * NN + k] * s_out[row] * INV_N;
        wl[i] = vv;
    }
    __syncthreads();

    const int wave = tid >> 5;
    const int lane = tid & 31;
    const int tile = blockIdx.x * 8 + wave;     // 500 blocks * 8 waves = 4000 tiles
    const int nCol = lane & 15;                 // A: M index ; B: N column
    const int koff = (lane < 16) ? 0 : 2;       // lanes 0-15: K{0,1}; 16-31: K{2,3}
    const int brow = (nCol < OUTC) ? nCol : OUTC;   // invalid columns -> zero row

    const float* hrow = H + (size_t)(tile * 16 + nCol) * NN + koff;
    const float* wrow = wl + (size_t)brow * NP + koff;

    v8f acc0 = {0.f,0.f,0.f,0.f,0.f,0.f,0.f,0.f};
    v8f acc1 = acc0, acc2 = acc0, acc3 = acc0;

    for (int kb = 0; kb < NN; kb += 16) {
        v2f a0 = *(const v2f*)(hrow + kb);
        v2f a1 = *(const v2f*)(hrow + kb + 4);
        v2f a2 = *(const v2f*)(hrow + kb + 8);
        v2f a3 = *(const v2f*)(hrow + kb + 12);
        const v2f b0 = *(const v2f*)(wrow + kb);
        const v2f b1 = *(const v2f*)(wrow + kb + 4);
        const v2f b2 = *(const v2f*)(wrow + kb + 8);
        const v2f b3 = *(const v2f*)(wrow + kb + 12);
        a0.x = my_tanh(a0.x); a0.y = my_tanh(a0.y);
        a1.x = my_tanh(a1.x); a1.y = my_tanh(a1.y);
        a2.x = my_tanh(a2.x); a2.y = my_tanh(a2.y);
        a3.x = my_tanh(a3.x); a3.y = my_tanh(a3.y);
        acc0 = __builtin_amdgcn_wmma_f32_16x16x4_f32(false, a0, false, b0,
                                                     (short)0, acc0, false, false);
        acc1 = __builtin_amdgcn_wmma_f32_16x16x4_f32(false, a1, false, b1,
                                                     (short)0, acc1, false, false);
        acc2 = __builtin_amdgcn_wmma_f32_16x16x4_f32(false, a2, false, b2,
                                                     (short)0, acc2, false, false);
        acc3 = __builtin_amdgcn_wmma_f32_16x16x4_f32(false, a3, false, b3,
                                                     (short)0, acc3, false, false);
    }

    v8f accT = (acc0 + acc1) + (acc2 + acc3);

    if (nCol < OUTC) {
        const int c = nCol;
        const int mofs = (lane < 16) ? 0 : 8;   // VGPR i <-> M = i (+8 upper lanes)
#pragma unroll
        for (int i = 0; i < 8; ++i)
            out[(size_t)(tile * 16 + i + mofs) * OUTC + c] = accT[i];
    }
}

// ================= launch =================
extern "C" void kernel_launch(void* const* d_in, const int* in_sizes, int n_in,
                              void* d_out, int out_size, void* d_ws, size_t ws_size,
                              hipStream_t stream) {
    (void)in_sizes; (void)n_in; (void)out_size; (void)d_ws; (void)ws_size;

    const float* X     = (const float*)d_in[0];  // [B,T,INC]
    const float* noise = (const float*)d_in[1];  // [B,T,N]
    const float* Win   = (const float*)d_in[2];  // [N,INC]
    const float* WV    = (const float*)d_in[3];  // [R,N]
    const float* WU    = (const float*)d_in[4];  // [N,R]
    const float* Wout  = (const float*)d_in[5];  // [OUTC,N]
    const float* s_in  = (const float*)d_in[6];  // [INC]
    const float* s_out = (const float*)d_in[7];  // [OUTC]
    const float* h0    = (const float*)d_in[8];  // [N]

    float* hidden = (float*)d_out;                        // [B,T,N]
    float* out    = hidden + (size_t)BB * TT * NN;        // [B,T,OUTC]

    rnn_scan_kernel<<<BB, 256, 0, stream>>>(X, noise, Win, WV, WU, s_in, h0, hidden);

    const int tiles  = (BB * TT) / 16;        // 4000
    const int blocks = tiles / 8;             // 500
    readout_wmma_kernel<<<blocks, 256, 0, stream>>>(hidden, Wout, s_out, out);
}